// QuanvolutionHybrid_65481071408909
// MI455X (gfx1250) — compile-verified
//
#include <hip/hip_runtime.h>
#include <hip/hip_bf16.h>
#include <math.h>

#define WIRES   4
#define NLAYERS 2
#define NCLS    10
#define BATCH   8192
#define PATCHES 196          // 14*14
#define FEATS   784          // PATCHES*WIRES
#define NPATCH  (BATCH * PATCHES)

typedef float v2f __attribute__((ext_vector_type(2)));
typedef float v8f __attribute__((ext_vector_type(8)));

// ---------------------------------------------------------------------------
// Kernel A: build the fixed 16x16 circuit matrix from params (8 scalars).
// Lane j simulates the layered circuit on basis state e_j and stores column j.
// Layout in workspace: Ut[j*16 + k] = U[k][j]  (column-major U == U^T rows).
// Wire w corresponds to bit (3-w) of the 4-bit amplitude index.
// ---------------------------------------------------------------------------
__global__ void build_unitary_kernel(const float* __restrict__ params,
                                     float* __restrict__ Ut) {
  const int lane = threadIdx.x;          // 32 launched; lanes 0..15 are used
  float st[16];
#pragma unroll
  for (int k = 0; k < 16; ++k) st[k] = (k == (lane & 15)) ? 1.0f : 0.0f;

#pragma unroll
  for (int l = 0; l < NLAYERS; ++l) {
    // Ry(params[l][w]) on each wire
#pragma unroll
    for (int w = 0; w < WIRES; ++w) {
      const float th = params[l * WIRES + w] * 0.5f;
      const float c = cosf(th), s = sinf(th);
      const int msk = 1 << (3 - w);
#pragma unroll
      for (int k = 0; k < 16; ++k) {
        if (!(k & msk)) {
          const float a0 = st[k], a1 = st[k | msk];
          st[k]       = c * a0 - s * a1;
          st[k | msk] = s * a0 + c * a1;
        }
      }
    }
    // CNOT ring: (0->1), (1->2), (2->3), (3->0)
    const int ctrls[4] = {0, 1, 2, 3};
    const int tgts[4]  = {1, 2, 3, 0};
#pragma unroll
    for (int e = 0; e < 4; ++e) {
      const int cm = 1 << (3 - ctrls[e]);
      const int tm = 1 << (3 - tgts[e]);
#pragma unroll
      for (int k = 0; k < 16; ++k) {
        if ((k & cm) && !(k & tm)) {
          const float t = st[k];
          st[k] = st[k | tm];
          st[k | tm] = t;
        }
      }
    }
  }
  if (lane < 16) {
#pragma unroll
    for (int k = 0; k < 16; ++k) Ut[lane * 16 + k] = st[k];
  }
}

// ---------------------------------------------------------------------------
// Kernel B: one thread per 2x2 patch.
// state = U * (product-state v built from 4 sincos pairs); z_w = sum(+-state^2)
// Feature vector (4 floats / patch) stored as one coalesced float4.
// ---------------------------------------------------------------------------
__global__ void __launch_bounds__(256)
quanv_kernel(const float* __restrict__ x,
             const float* __restrict__ Ut,
             float* __restrict__ feats) {
  __shared__ float Us[256];
  const int tid = threadIdx.x;
  Us[tid] = Ut[tid];                     // 256 threads load 256 floats
  __syncthreads();

  const int p = blockIdx.x * 256 + tid;
  if (p >= NPATCH) return;

  const int b     = p / PATCHES;
  const int patch = p - b * PATCHES;
  const int r     = patch / 14;
  const int c     = patch - r * 14;

  const float* px = x + ((size_t)(b * 28 + 2 * r) * 28 + 2 * c);
  const float a00 = px[0], a01 = px[1], a10 = px[28], a11 = px[29];

  float s0, c0, s1, c1, s2, c2, s3, c3;
  __sincosf(a00 * 0.5f, &s0, &c0);   // wire 0 (bit 3)
  __sincosf(a01 * 0.5f, &s1, &c1);   // wire 1 (bit 2)
  __sincosf(a10 * 0.5f, &s2, &c2);   // wire 2 (bit 1)
  __sincosf(a11 * 0.5f, &s3, &c3);   // wire 3 (bit 0)

  const float t01[4] = {c0 * c1, c0 * s1, s0 * c1, s0 * s1};
  const float t23[4] = {c2 * c3, c2 * s3, s2 * c3, s2 * s3};

  float st[16];
#pragma unroll
  for (int k = 0; k < 16; ++k) st[k] = 0.0f;

#pragma unroll
  for (int j = 0; j < 16; ++j) {
    const float vj = t01[j >> 2] * t23[j & 3];
#pragma unroll
    for (int k = 0; k < 16; ++k)
      st[k] = __fmaf_rn(Us[j * 16 + k], vj, st[k]);   // LDS broadcast reads
  }

  float pr[16];
#pragma unroll
  for (int k = 0; k < 16; ++k) pr[k] = st[k] * st[k];

  float z[4];
#pragma unroll
  for (int w = 0; w < 4; ++w) {
    const int msk = 8 >> w;
    float acc = 0.0f;
#pragma unroll
    for (int k = 0; k < 16; ++k) acc += (k & msk) ? -pr[k] : pr[k];
    z[w] = acc;
  }

  reinterpret_cast<float4*>(feats)[p] = make_float4(z[0], z[1], z[2], z[3]);
}

// ---------------------------------------------------------------------------
// Kernel C: logits = feats @ W^T + b, then row-wise log_softmax.
// fp32 WMMA 16x16x4: each wave owns a 16(batch) x 16(classes, padded) tile.
// A frag (16x4 f32): lane&15 = M row, lane>>4 selects K-pair {0,1}/{2,3}.
// B frag (4x16 f32): lane&15 = N col, lane>>4 selects K-pair; W kept
// transposed in LDS (Wt[k*16+n]) so lane-consecutive reads hit distinct banks.
// C/D (16x16 f32, 8 VGPRs): acc[r] = logits[m0 + r + 8*(lane>>4)][lane&15].
// log-softmax reductions use shfl_xor masks 1,2,4,8 (stay inside 16-lane
// halves of the wave32, which hold independent output rows).
// ---------------------------------------------------------------------------
__global__ void __launch_bounds__(256)
classifier_kernel(const float* __restrict__ feats,
                  const float* __restrict__ W,
                  const float* __restrict__ bias,
                  float* __restrict__ out) {
  __shared__ float Wt[FEATS * 16];       // 50,176 B
  const int tid = threadIdx.x;
  for (int i = tid; i < FEATS * 16; i += 256) {
    const int k = i >> 4, n = i & 15;
    Wt[i] = (n < NCLS) ? W[n * FEATS + k] : 0.0f;
  }
  __syncthreads();

  const int wave = tid >> 5;
  const int lane = tid & 31;
  const int n    = lane & 15;            // N column / A row
  const int kh   = lane >> 4;            // which K pair this lane holds
  const int m0   = (blockIdx.x * 8 + wave) * 16;

  const float* arow = feats + (size_t)(m0 + n) * FEATS + 2 * kh;

  v8f acc = {0.f, 0.f, 0.f, 0.f, 0.f, 0.f, 0.f, 0.f};

  for (int k0 = 0; k0 < FEATS; k0 += 16) {
    __builtin_prefetch(arow + k0 + 256, 0, 1);   // global_prefetch_b8
#pragma unroll
    for (int kk = 0; kk < 16; kk += 4) {
      const int k = k0 + kk;
      const v2f a = *(const v2f*)(arow + k);     // 8B-aligned global load
      v2f bb;
      bb.x = Wt[(k + 2 * kh) * 16 + n];
      bb.y = Wt[(k + 2 * kh + 1) * 16 + n];
      acc = __builtin_amdgcn_wmma_f32_16x16x4_f32(
          /*neg_a=*/false, a, /*neg_b=*/false, bb,
          /*c_mod=*/(short)0, acc, /*reuse_a=*/false, /*reuse_b=*/false);
    }
  }

  const float bn = (n < NCLS) ? bias[n] : 0.0f;
#pragma unroll
  for (int r = 0; r < 8; ++r) {
    const int row = m0 + r + 8 * kh;
    float val = (n < NCLS) ? (acc[r] + bn) : -3.0e38f;
    float mx = val;
#pragma unroll
    for (int msk = 1; msk <= 8; msk <<= 1)
      mx = fmaxf(mx, __shfl_xor(mx, msk, 32));
    float e = (n < NCLS) ? __expf(val - mx) : 0.0f;
    float sum = e;
#pragma unroll
    for (int msk = 1; msk <= 8; msk <<= 1)
      sum += __shfl_xor(sum, msk, 32);
    const float res = val - mx - __logf(sum);
    if (n < NCLS) out[(size_t)row * NCLS + n] = res;
  }
}

// ---------------------------------------------------------------------------
extern "C" void kernel_launch(void* const* d_in, const int* in_sizes, int n_in,
                              void* d_out, int out_size, void* d_ws, size_t ws_size,
                              hipStream_t stream) {
  const float* x      = (const float*)d_in[0];   // [8192,28,28]
  const float* params = (const float*)d_in[1];   // [2,4]
  const float* W      = (const float*)d_in[2];   // [10,784]
  const float* bias   = (const float*)d_in[3];   // [10]
  float* out = (float*)d_out;                    // [8192,10]

  float* Ut    = (float*)d_ws;                   // 256 floats (1 KB)
  float* feats = (float*)d_ws + 256;             // 8192*784 floats (~25.7 MB)

  build_unitary_kernel<<<1, 32, 0, stream>>>(params, Ut);
  quanv_kernel<<<(NPATCH + 255) / 256, 256, 0, stream>>>(x, Ut, feats);
  classifier_kernel<<<BATCH / 128, 256, 0, stream>>>(feats, W, bias, out);
}